// TCN_GCN_unit_1228360647386
// MI455X (gfx1250) — compile-verified
//
#include <hip/hip_runtime.h>
#include <math.h>

// ---------------------------------------------------------------------------
// Problem constants (from the reference):
//   x : (N=64, C=256, T=64, V=50) f32, row-major, v fastest
//   W1: (10, H=16, C=256)  b1: (10,16)   W2: (10, C=256, H=16)  b2: (10,256)
//   out: (64, 256, 64, 50) f32, V axis re-ordered by part concatenation
// Memory-bound: ~630 MB total traffic => ~27us at 23.3 TB/s. MLP ~21 MFLOP,
// done with v_wmma_f32_16x16x4_f32 (full f32 precision, matrix pipe).
// ---------------------------------------------------------------------------

#define N_BATCH 64
#define C_DIM   256
#define T_DIM   64
#define V_DIM   50
#define H_DIM   16
#define NPARTS  10

typedef __attribute__((ext_vector_type(2))) float v2f;
typedef __attribute__((ext_vector_type(8))) float v8f;

// part id for each original vertex v (PARTS from the reference)
__constant__ int d_part_of_v[V_DIM] = {
    0,0,0,0,  3,3,3,3,  1,1,1,1,  4,4,4,4,  2,2,2,2,  0, 3,3, 1,1,
    5,5,5,5,  8,8,8,8,  6,6,6,6,  9,9,9,9,  7,7,7,7,  5, 8,8, 6,6
};
// elements per part (for the mean)
__constant__ int d_part_len[NPARTS] = {5,6,4,6,4,5,6,4,6,4};
// output column j -> source vertex (concatenation order of PARTS)
__constant__ int d_src_v[V_DIM] = {
    0,1,2,3,20,           // p0
    8,9,10,11,23,24,      // p1
    16,17,18,19,          // p2
    4,5,6,7,21,22,        // p3
    12,13,14,15,          // p4
    25,26,27,28,45,       // p5
    33,34,35,36,48,49,    // p6
    41,42,43,44,          // p7
    29,30,31,32,46,47,    // p8
    37,38,39,40           // p9
};
// output column j -> part id
__constant__ int d_part_of_out[V_DIM] = {
    0,0,0,0,0, 1,1,1,1,1,1, 2,2,2,2, 3,3,3,3,3,3, 4,4,4,4,
    5,5,5,5,5, 6,6,6,6,6,6, 7,7,7,7, 8,8,8,8,8,8, 9,9,9,9
};

// Order-preserving key for float -> unsigned max (works for all finite floats)
__device__ __forceinline__ unsigned fkey(float f) {
    unsigned u = __float_as_uint(f);
    return (u & 0x80000000u) ? ~u : (u | 0x80000000u);
}
__device__ __forceinline__ float fdec(unsigned k) {
    unsigned u = (k & 0x80000000u) ? (k ^ 0x80000000u) : ~k;
    return __uint_as_float(u);
}

// ---------------------------------------------------------------------------
// Pass 1: per-(n,c) per-part sum & max over (T x part-vertices).
// One block per (n,c): 3200 contiguous floats. Thread tid<200 owns vertex
// v=tid%50 (fixed part => private register accumulation) over 16 t-steps.
// Cross-thread combine via LDS float-add / u32-max atomics (ds_add_f32,
// ds_max_u32): 200 atomics per block, negligible vs HBM latency.
// ---------------------------------------------------------------------------
__global__ __launch_bounds__(256) void stats_kernel(const float* __restrict__ x,
                                                    float* __restrict__ avg_buf,
                                                    float* __restrict__ max_buf) {
    __shared__ float    s_sum[NPARTS];
    __shared__ unsigned s_maxk[NPARTS];
    const int b   = blockIdx.x;          // b = n*C + c
    const int tid = threadIdx.x;

    if (tid < NPARTS) { s_sum[tid] = 0.0f; s_maxk[tid] = 0u; }
    __syncthreads();

    if (tid < 4 * V_DIM) {
        const int v  = tid % V_DIM;
        const int tg = tid / V_DIM;                       // 0..3, 16 t each
        const int p  = d_part_of_v[v];
        const float* row = x + (size_t)b * (T_DIM * V_DIM) + tg * 16 * V_DIM + v;
        float s = 0.0f;
        float m = -__builtin_huge_valf();
#pragma unroll
        for (int t = 0; t < 16; ++t) {
            float val = row[t * V_DIM];
            s += val;
            m  = fmaxf(m, val);
        }
        atomicAdd(&s_sum[p], s);
        atomicMax(&s_maxk[p], fkey(m));
    }
    __syncthreads();

    if (tid < NPARTS) {
        const int p = tid;
        const size_t o = (size_t)p * (N_BATCH * C_DIM) + b;   // [p][n][c]
        avg_buf[o] = s_sum[p] / (float)(T_DIM * d_part_len[p]);
        max_buf[o] = fdec(s_maxk[p]);
    }
}

// ---------------------------------------------------------------------------
// Pass 2: gate(p,n,c) = sigmoid( mlp(avg) + mlp(mx) ) via f32 WMMA.
// One block per part, 4 waves (wave32). Wave w owns batch rows 16w..16w+15
// and carries BOTH avg & mx accumulators so each B fragment is loaded once.
//
// Layouts per ISA 7.12.2 (wave32):
//   A 16x4 f32 : lane = M (lo/hi 16-lane halves give K-pairs), 2 VGPRs
//                lane L, reg r holds A[L%16][2*(L>>4)+r]
//   B 4x16 f32 : lane = N, reg r holds B[2*(L>>4)+r][L%16]
//   C/D 16x16  : reg i, lane L holds D[i + 8*(L>>4)][L%16]
// Layer-2 A comes from layer-1 D => C-layout -> A-layout transpose via LDS.
// ---------------------------------------------------------------------------
__global__ __launch_bounds__(128) void gate_kernel(const float* __restrict__ avg_buf,
                                                   const float* __restrict__ max_buf,
                                                   const float* __restrict__ W1,
                                                   const float* __restrict__ b1,
                                                   const float* __restrict__ W2,
                                                   const float* __restrict__ b2,
                                                   float* __restrict__ gate_buf) {
    const int p    = blockIdx.x;
    const int tid  = threadIdx.x;
    const int wave = tid >> 5;       // 0..3 : 16-row batch tile
    const int lane = tid & 31;
    const int l16  = lane & 15;
    const int hi   = lane >> 4;      // 0/1 : K-pair selector for A/B frags

    const float* Ap  = avg_buf + (size_t)p * (N_BATCH * C_DIM);
    const float* Mp  = max_buf + (size_t)p * (N_BATCH * C_DIM);
    const float* W1p = W1 + (size_t)p * (H_DIM * C_DIM);   // [h][c]
    const float* W2p = W2 + (size_t)p * (C_DIM * H_DIM);   // [c][h]

    const int row = wave * 16 + l16;                       // batch row for A

    // ---- Layer 1: h(128x16) = v(128x256) @ W1^T, K=256 in steps of 4 ----
    v8f acc_a = {};
    v8f acc_m = {};
    for (int k0 = 0; k0 < C_DIM; k0 += 4) {
        const int k = k0 + 2 * hi;
        // B[k][h] = W1^T[k][h] = W1p[h*256 + k]; pair (k,k+1) is contiguous
        v2f bfrag = *(const v2f*)(W1p + l16 * C_DIM + k);
        v2f aa    = *(const v2f*)(Ap + row * C_DIM + k);
        v2f am    = *(const v2f*)(Mp + row * C_DIM + k);
        acc_a = __builtin_amdgcn_wmma_f32_16x16x4_f32(false, aa, false, bfrag,
                                                      (short)0, acc_a, false, false);
        acc_m = __builtin_amdgcn_wmma_f32_16x16x4_f32(false, am, false, bfrag,
                                                      (short)0, acc_m, false, false);
    }

    // bias + ReLU; stage h tiles in LDS to convert C-layout -> A-layout
    __shared__ __align__(16) float h_lds[4][2][16][16];    // [wave][avg/mx][m][h]
    const float b1v = b1[p * H_DIM + l16];
#pragma unroll
    for (int i = 0; i < 8; ++i) {
        const int m = i + 8 * hi;
        h_lds[wave][0][m][l16] = fmaxf(acc_a[i] + b1v, 0.0f);
        h_lds[wave][1][m][l16] = fmaxf(acc_m[i] + b1v, 0.0f);
    }
    __syncthreads();   // per-wave data, but keep all lanes converged & ordered

    // A2 fragments for layer 2: lane L, k-step t needs h[L%16][4t + 2*hi (+1)]
    v2f a2a[4], a2m[4];
#pragma unroll
    for (int t = 0; t < 4; ++t) {
        const int k = 4 * t + 2 * hi;
        a2a[t] = *(const v2f*)&h_lds[wave][0][l16][k];
        a2m[t] = *(const v2f*)&h_lds[wave][1][l16][k];
    }

    // ---- Layer 2: g(128x256) = h @ W2^T, 16 column tiles, K=16 ----
    for (int j = 0; j < 16; ++j) {
        const int c = j * 16 + l16;                        // output channel
        v8f ga = {};
        v8f gm = {};
#pragma unroll
        for (int t = 0; t < 4; ++t) {
            const int k = 4 * t + 2 * hi;
            // B[k][c] = W2^T[k][c] = W2p[c*16 + k]; pair contiguous
            v2f b2f = *(const v2f*)(W2p + c * H_DIM + k);
            ga = __builtin_amdgcn_wmma_f32_16x16x4_f32(false, a2a[t], false, b2f,
                                                       (short)0, ga, false, false);
            gm = __builtin_amdgcn_wmma_f32_16x16x4_f32(false, a2m[t], false, b2f,
                                                       (short)0, gm, false, false);
        }
        // mlp(avg)+mlp(mx) includes b2 twice; then sigmoid
        const float bb = 2.0f * b2[p * C_DIM + c];
#pragma unroll
        for (int i = 0; i < 8; ++i) {
            const int n = wave * 16 + i + 8 * hi;          // batch index
            const float s = ga[i] + gm[i] + bb;
            const float g = 1.0f / (1.0f + __expf(-s));
            gate_buf[((size_t)p * N_BATCH + n) * C_DIM + c] = g;
        }
    }
}

// ---------------------------------------------------------------------------
// Pass 3: out[n][c][t][j] = x[n][c][t][src_v[j]] * gate[part(j)][n][c].
// One block per (n,c); stores are fully coalesced B32, gathered reads stay
// inside the same 200B row window. Gather tables + gates staged in LDS.
// ---------------------------------------------------------------------------
__global__ __launch_bounds__(256) void apply_kernel(const float* __restrict__ x,
                                                    const float* __restrict__ gate_buf,
                                                    float* __restrict__ out) {
    const int b   = blockIdx.x;          // n*C + c
    const int tid = threadIdx.x;

    __shared__ float s_gv[V_DIM];        // gate per OUTPUT column
    __shared__ int   s_src[V_DIM];       // source vertex per output column
    if (tid < V_DIM) {
        const int pj = d_part_of_out[tid];
        s_gv[tid]  = gate_buf[(size_t)pj * (N_BATCH * C_DIM) + b];
        s_src[tid] = d_src_v[tid];
    }
    __syncthreads();

    const size_t base = (size_t)b * (T_DIM * V_DIM);
    const float* xr  = x   + base;
    float*       orw = out + base;
    for (int i = tid; i < T_DIM * V_DIM; i += 256) {
        const int t = i / V_DIM;
        const int j = i - t * V_DIM;
        orw[i] = xr[t * V_DIM + s_src[j]] * s_gv[j];
    }
}

// ---------------------------------------------------------------------------
extern "C" void kernel_launch(void* const* d_in, const int* in_sizes, int n_in,
                              void* d_out, int out_size, void* d_ws, size_t ws_size,
                              hipStream_t stream) {
    const float* x  = (const float*)d_in[0];
    const float* W1 = (const float*)d_in[1];
    const float* b1 = (const float*)d_in[2];
    const float* W2 = (const float*)d_in[3];
    const float* b2 = (const float*)d_in[4];
    float* out = (float*)d_out;

    // workspace: avg | max | gate, each (10,64,256) f32  => ~2 MB total
    float* avg_buf  = (float*)d_ws;
    float* max_buf  = avg_buf + NPARTS * N_BATCH * C_DIM;
    float* gate_buf = max_buf + NPARTS * N_BATCH * C_DIM;

    const dim3 gridNC(N_BATCH * C_DIM);
    stats_kernel<<<gridNC, 256, 0, stream>>>(x, avg_buf, max_buf);
    gate_kernel<<<NPARTS, 128, 0, stream>>>(avg_buf, max_buf, W1, b1, W2, b2, gate_buf);
    apply_kernel<<<gridNC, 256, 0, stream>>>(x, gate_buf, out);
}